// AxisLengthNetMetric_76390288327516
// MI455X (gfx1250) — compile-verified
//
#include <hip/hip_runtime.h>
#include <math.h>

typedef __attribute__((ext_vector_type(2))) float v2f;
typedef __attribute__((ext_vector_type(8))) float v8f;

#define THREADS 256
#define BLOCKS  1024

// DPP ctrl encodings
#define DPP_QUAD_XOR1   0xB1   // quad_perm(1,0,3,2)
#define DPP_QUAD_XOR2   0x4E   // quad_perm(2,3,0,1)
#define DPP_HALF_MIRROR 0x141  // lane -> lane^7 within 8-lane half-row

template <int CTRL>
__device__ __forceinline__ float dpp_min(float x) {
    int xi = __float_as_int(x);
    // old = xi, bound_ctrl = false: safe regardless of inactive lanes
    int yi = __builtin_amdgcn_update_dpp(xi, xi, CTRL, 0xF, 0xF, false);
    return fminf(x, __int_as_float(yi));
}

__global__ __launch_bounds__(THREADS)
void chamfer_main(const float* __restrict__ pred_pts,   // (N,8,3)
                  const float* __restrict__ pred_size,  // (N,3)
                  const float* __restrict__ gt_box,     // (N,12)
                  float* __restrict__ partials,         // (BLOCKS,2)
                  int nBoxes)
{
    const int lane         = threadIdx.x & 31;
    const int waveInBlock  = threadIdx.x >> 5;
    const int wavesPerBlk  = THREADS / 32;
    const int waveGlobal   = blockIdx.x * wavesPerBlk + waveInBlock;
    const int totalWaves   = gridDim.x * wavesPerBlk;
    const int nPairs       = nBoxes >> 1;

    const int  half = lane & 15;   // row/col index in the 16x16 tile
    const int  sub  = half >> 3;   // which box of the pair
    const int  c    = half & 7;    // corner / pred-point index
    const bool hi   = lane >= 16;  // upper half-wave carries K=2,3

    float cdAcc = 0.0f;
    float l1Acc = 0.0f;

    for (int pair = waveGlobal; pair < nPairs; pair += totalWaves) {
        const int box = pair * 2 + sub;

        // ---- gt record: size[3:6], rx[6:9], ry[9:12] ----
        const float* g = gt_box + (size_t)box * 12;
        float s0 = g[3], s1 = g[4], s2 = g[5];
        float rx0 = g[6], rx1 = g[7], rx2 = g[8];
        float ry0 = g[9], ry1 = g[10], ry2 = g[11];
        float inx = 1.0f / (sqrtf(rx0*rx0 + rx1*rx1 + rx2*rx2) + 1e-8f);
        float iny = 1.0f / (sqrtf(ry0*ry0 + ry1*ry1 + ry2*ry2) + 1e-8f);
        rx0 *= inx; rx1 *= inx; rx2 *= inx;
        ry0 *= iny; ry1 *= iny; ry2 *= iny;
        float rz0 = rx1*ry2 - rx2*ry1;
        float rz1 = rx2*ry0 - rx0*ry2;
        float rz2 = rx0*ry1 - rx1*ry0;
        // SIGNS bit order: sx=bit2, sy=bit1, sz=bit0
        float fx = (c & 4) ? 0.5f*s0 : -0.5f*s0;
        float fy = (c & 2) ? 0.5f*s1 : -0.5f*s1;
        float fz = (c & 1) ? 0.5f*s2 : -0.5f*s2;
        float cx = fx*rx0 + fy*ry0 + fz*rz0;
        float cy = fx*rx1 + fy*ry1 + fz*rz1;
        float cz = fx*rx2 + fy*ry2 + fz*rz2;
        float a2 = cx*cx + cy*cy + cz*cz;

        // ---- A fragment: row = (cx, cy, cz, a2) ----
        v2f a;
        a.x = hi ? cz : cx;
        a.y = hi ? a2 : cy;

        // ---- B fragment: col = (-2px, -2py, -2pz, 1) ----
        const float* p = pred_pts + ((size_t)box * 8 + c) * 3;
        float px = p[0], py = p[1], pz = p[2];
        float b2 = px*px + py*py + pz*pz;
        v2f b;
        b.x = hi ? -2.0f*pz : -2.0f*px;
        b.y = hi ? 1.0f    : -2.0f*py;

        // ---- WMMA: D[p][q] = a2[p] - 2 * corner_p . pred_q ----
        v8f acc = {};
        acc = __builtin_amdgcn_wmma_f32_16x16x4_f32(false, a, false, b,
                                                    (short)0, acc, false, false);

        // d[p][q] = max(a2 + b2 - 2ab, 0); lane owns column q=half, rows (hi?8:0)+v
        float d[8];
        #pragma unroll
        for (int v = 0; v < 8; ++v) {
            float dv = acc[v] + b2;
            d[v] = dv > 0.0f ? dv : 0.0f;
        }
        // dist2[q]: min over the 8 rows in this lane
        float colmin = d[0];
        #pragma unroll
        for (int v = 1; v < 8; ++v) colmin = fminf(colmin, d[v]);
        // dist1[row]: min across the 8 lanes of the group, pure DPP (no LDS)
        #pragma unroll
        for (int v = 0; v < 8; ++v) {
            float t = d[v];
            t = dpp_min<DPP_QUAD_XOR1>(t);
            t = dpp_min<DPP_QUAD_XOR2>(t);      // quad-uniform now
            t = dpp_min<DPP_HALF_MIRROR>(t);    // xor7 == xor4 partner's value
            d[v] = t;
        }
        // useful diagonal blocks: lanes 0-7 (box0) and 24-31 (box1)
        bool useful = (lane < 8) || (lane >= 24);
        if (useful) cdAcc += colmin + d[lane & 7];

        // ---- L1 size term: lanes 0..2 -> box0, lanes 8..10 -> box1 ----
        if (!hi && c < 3) {
            float s = (c == 0) ? s0 : (c == 1) ? s1 : s2;
            l1Acc += fabsf(s - pred_size[(size_t)box * 3 + c]);
        }
    }

    // ---- block reduction ----
    __shared__ float sCd[THREADS];
    __shared__ float sL1[THREADS];
    sCd[threadIdx.x] = cdAcc;
    sL1[threadIdx.x] = l1Acc;
    __syncthreads();
    for (int s = THREADS / 2; s > 0; s >>= 1) {
        if (threadIdx.x < (unsigned)s) {
            sCd[threadIdx.x] += sCd[threadIdx.x + s];
            sL1[threadIdx.x] += sL1[threadIdx.x + s];
        }
        __syncthreads();
    }
    if (threadIdx.x == 0) {
        partials[blockIdx.x * 2 + 0] = sCd[0];
        partials[blockIdx.x * 2 + 1] = sL1[0];
    }
}

__global__ void chamfer_final(const float* __restrict__ partials,
                              float* __restrict__ out,
                              int nBlocks, int nBoxes)
{
    if (threadIdx.x == 0 && blockIdx.x == 0) {
        double cd = 0.0, l1 = 0.0;
        for (int i = 0; i < nBlocks; ++i) {   // fixed order -> deterministic
            cd += (double)partials[2 * i + 0];
            l1 += (double)partials[2 * i + 1];
        }
        out[0] = (float)(cd / ((double)nBoxes * 8.0));
        out[1] = (float)(l1 / ((double)nBoxes * 3.0));
    }
}

extern "C" void kernel_launch(void* const* d_in, const int* in_sizes, int n_in,
                              void* d_out, int out_size, void* d_ws, size_t ws_size,
                              hipStream_t stream) {
    const float* pred_pts  = (const float*)d_in[0];
    const float* pred_size = (const float*)d_in[1];
    const float* gt_box    = (const float*)d_in[2];
    int nBoxes = in_sizes[1] / 3;           // (N,3) -> N
    float* partials = (float*)d_ws;         // 2*BLOCKS floats, overwritten each call

    chamfer_main<<<BLOCKS, THREADS, 0, stream>>>(pred_pts, pred_size, gt_box,
                                                 partials, nBoxes);
    chamfer_final<<<1, 32, 0, stream>>>(partials, (float*)d_out, BLOCKS, nBoxes);
}